// GCNWithAttention_81527069212874
// MI455X (gfx1250) — compile-verified
//
#include <hip/hip_runtime.h>
#include <math.h>

typedef __attribute__((ext_vector_type(2))) float v2f;
typedef __attribute__((ext_vector_type(8))) float v8f;

#define LDSS 132   // 16x128 A-tile LDS stride (132 % 64 = 4 -> conflict-free row access)

// ---------------- utility ----------------
__global__ void zero_kernel(float* __restrict__ p, int n) {
  int i = blockIdx.x * blockDim.x + threadIdx.x;
  int stride = gridDim.x * blockDim.x;
  for (; i < n; i += stride) p[i] = 0.0f;
}

__global__ void degree_kernel(const int* __restrict__ dst, float* __restrict__ deg, int E) {
  int i = blockIdx.x * blockDim.x + threadIdx.x;
  if (i < E) atomicAdd(&deg[dst[i]], 1.0f);
}

__global__ void rsqrt_kernel(float* __restrict__ d, int N) {
  int i = blockIdx.x * blockDim.x + threadIdx.x;
  if (i < N) d[i] = rsqrtf(d[i] + 1.0f);   // +1 self-loop
}

// ---------------- WMMA fp32 GEMM:  Y[M,128] = act(X[M,128] @ W[128,128] + bias) ----------------
// Block = 256 threads = 8 waves; block owns one 16-row M tile; wave w owns N-cols [16w,16w+16).
// A fragment (16x4 f32): lanes 0-15 -> (M=lane, K=k..k+1), lanes 16-31 -> (M=lane-16, K=k+2..k+3).
// B fragment (4x16 f32): v0 = B[k][n]/B[k+2][n], v1 = B[k+1][n]/B[k+3][n] (mirrored layout).
// C/D: VGPR i -> rows i (lanes 0-15) and i+8 (lanes 16-31).
__global__ void wmma_gemm_k128(const float* __restrict__ X, const float* __restrict__ W,
                               const float* __restrict__ bias, float* __restrict__ Y,
                               int M, int do_relu) {
  __shared__ float Xs[16 * LDSS];
  const int tid = threadIdx.x;
  const int m0  = blockIdx.x * 16;

  for (int i = tid; i < 16 * 128; i += 256) {
    int row = i >> 7, col = i & 127;
    int gr = m0 + row; if (gr >= M) gr = M - 1;
    Xs[row * LDSS + col] = X[(size_t)gr * 128 + col];
  }
  __syncthreads();

  const int wave = tid >> 5;
  const int lane = tid & 31;
  const int n0   = wave * 16;
  const int half = lane >> 4;     // 0: K lo pair, 1: K hi pair
  const int r    = lane & 15;

  v8f acc = {};
  for (int k = 0; k < 128; k += 4) {
    const int kb = k + (half ? 2 : 0);
    v2f a, b;
    a[0] = Xs[r * LDSS + kb];
    a[1] = Xs[r * LDSS + kb + 1];
    b[0] = W[(size_t)kb * 128 + n0 + r];
    b[1] = W[(size_t)(kb + 1) * 128 + n0 + r];
    acc = __builtin_amdgcn_wmma_f32_16x16x4_f32(false, a, false, b,
                                                (short)0, acc, false, false);
  }

  const float bv = bias ? bias[n0 + r] : 0.0f;
  const int rowbase = m0 + (half ? 8 : 0);
  for (int i = 0; i < 8; ++i) {
    int gr = rowbase + i;
    if (gr < M) {
      float v = acc[i] + bv;
      if (do_relu) v = fmaxf(v, 0.0f);
      Y[(size_t)gr * 128 + n0 + r] = v;
    }
  }
}

// ---------------- edge scatter: agg[dst] += H[src] * dis[src]*dis[dst] ----------------
__global__ void mp_scatter(const float* __restrict__ H, const int* __restrict__ src,
                           const int* __restrict__ dst, const float* __restrict__ dis,
                           float* __restrict__ agg, int E) {
  const int lane = threadIdx.x & 31;
  const int wstride = gridDim.x * 8;
  for (int e = blockIdx.x * 8 + (threadIdx.x >> 5); e < E; e += wstride) {
    const int s = src[e], d = dst[e];
    const float w = dis[s] * dis[d];
    const float4* hp = (const float4*)(H + (size_t)s * 128);
    float4 v = hp[lane];
    float* ap = agg + (size_t)d * 128 + lane * 4;
    atomicAdd(ap + 0, v.x * w);
    atomicAdd(ap + 1, v.y * w);
    atomicAdd(ap + 2, v.z * w);
    atomicAdd(ap + 3, v.w * w);
  }
}

// ---------------- self-loop + bias + optional relu (in place on agg) ----------------
__global__ void self_bias_act(float* __restrict__ agg, const float* __restrict__ H,
                              const float* __restrict__ dis, const float* __restrict__ b,
                              int total, int do_relu) {
  int i = blockIdx.x * blockDim.x + threadIdx.x;
  if (i >= total) return;
  int n = i >> 7, f = i & 127;
  float di = dis[n];
  float v = agg[i] + H[i] * di * di + b[f];
  agg[i] = do_relu ? fmaxf(v, 0.0f) : v;
}

// ---------------- attention logits: a[n] = leaky_relu(h[n].Wa + ba) ----------------
__global__ void attn_logits(const float* __restrict__ H, const float* __restrict__ Wa,
                            const float* __restrict__ ba, float* __restrict__ a, int N) {
  const int lane = threadIdx.x & 31;
  const int n = blockIdx.x * 8 + (threadIdx.x >> 5);
  if (n >= N) return;
  const float4* hp = (const float4*)(H + (size_t)n * 128);
  const float4* wp = (const float4*)Wa;
  float4 h = hp[lane], w = wp[lane];
  float d = h.x * w.x + h.y * w.y + h.z * w.z + h.w * w.w;
  for (int off = 16; off; off >>= 1) d += __shfl_down(d, off, 32);
  if (lane == 0) {
    float v = d + ba[0];
    a[n] = v > 0.0f ? v : 0.01f * v;   // jax leaky_relu default slope
  }
}

// ---------------- global softmax stats over all nodes (single block) ----------------
__global__ void softmax_stats(const float* __restrict__ a, float* __restrict__ stats, int N) {
  __shared__ float red[1024];
  const int tid = threadIdx.x;
  float m = -3.4e38f;
  for (int i = tid; i < N; i += 1024) m = fmaxf(m, a[i]);
  red[tid] = m; __syncthreads();
  for (int s = 512; s; s >>= 1) { if (tid < s) red[tid] = fmaxf(red[tid], red[tid + s]); __syncthreads(); }
  const float mx = red[0]; __syncthreads();
  float sum = 0.0f;
  for (int i = tid; i < N; i += 1024) sum += expf(a[i] - mx);
  red[tid] = sum; __syncthreads();
  for (int s = 512; s; s >>= 1) { if (tid < s) red[tid] += red[tid + s]; __syncthreads(); }
  if (tid == 0) { stats[0] = mx; stats[1] = red[0]; }
}

// ---------------- pooled[batch[n]] += h[n] * softmax(a)[n]; cnt[g] += 1 ----------------
__global__ void pool_scatter(const float* __restrict__ H, const float* __restrict__ a,
                             const float* __restrict__ stats, const int* __restrict__ batch,
                             float* __restrict__ pooled, float* __restrict__ cnt, int N) {
  const int lane = threadIdx.x & 31;
  const int n = blockIdx.x * 8 + (threadIdx.x >> 5);
  if (n >= N) return;
  const float w = expf(a[n] - stats[0]) / stats[1];
  const int g = batch[n];
  const float4* hp = (const float4*)(H + (size_t)n * 128);
  float4 v = hp[lane];
  float* pp = pooled + (size_t)g * 128 + lane * 4;
  atomicAdd(pp + 0, v.x * w);
  atomicAdd(pp + 1, v.y * w);
  atomicAdd(pp + 2, v.z * w);
  atomicAdd(pp + 3, v.w * w);
  if (lane == 0) atomicAdd(&cnt[g], 1.0f);
}

__global__ void pooled_div(float* __restrict__ pooled, const float* __restrict__ cnt, int total) {
  int i = blockIdx.x * blockDim.x + threadIdx.x;
  if (i < total) pooled[i] /= fmaxf(cnt[i >> 7], 1.0f);
}

// ---------------- head: out[g] = softmax(z2[g] @ Wo[128,2] + bo) ----------------
__global__ void head_kernel(const float* __restrict__ Z, const float* __restrict__ Wo,
                            const float* __restrict__ bo, float* __restrict__ out, int G) {
  const int lane = threadIdx.x & 31;
  const int g = blockIdx.x * 8 + (threadIdx.x >> 5);
  if (g >= G) return;
  const float4* zp = (const float4*)(Z + (size_t)g * 128);
  float4 z = zp[lane];
  const int k = lane * 4;
  float a0 = z.x * Wo[(k + 0) * 2] + z.y * Wo[(k + 1) * 2] + z.z * Wo[(k + 2) * 2] + z.w * Wo[(k + 3) * 2];
  float a1 = z.x * Wo[(k + 0) * 2 + 1] + z.y * Wo[(k + 1) * 2 + 1] + z.z * Wo[(k + 2) * 2 + 1] + z.w * Wo[(k + 3) * 2 + 1];
  for (int off = 16; off; off >>= 1) {
    a0 += __shfl_down(a0, off, 32);
    a1 += __shfl_down(a1, off, 32);
  }
  if (lane == 0) {
    float l0 = a0 + bo[0], l1 = a1 + bo[1];
    float m = fmaxf(l0, l1);
    float e0 = expf(l0 - m), e1 = expf(l1 - m);
    float inv = 1.0f / (e0 + e1);
    out[g * 2 + 0] = e0 * inv;
    out[g * 2 + 1] = e1 * inv;
  }
}

extern "C" void kernel_launch(void* const* d_in, const int* in_sizes, int n_in,
                              void* d_out, int out_size, void* d_ws, size_t ws_size,
                              hipStream_t stream) {
  (void)n_in; (void)ws_size;
  const float* x    = (const float*)d_in[0];
  const int*   ei   = (const int*)d_in[1];
  const int*   batch= (const int*)d_in[2];
  const float* W1   = (const float*)d_in[3];
  const float* b1   = (const float*)d_in[4];
  const float* W2   = (const float*)d_in[5];
  const float* b2   = (const float*)d_in[6];
  const float* Wa   = (const float*)d_in[7];
  const float* ba   = (const float*)d_in[8];
  const float* Wl1  = (const float*)d_in[9];
  const float* bl1  = (const float*)d_in[10];
  const float* Wl2  = (const float*)d_in[11];
  const float* bl2  = (const float*)d_in[12];
  const float* Wo   = (const float*)d_in[13];
  const float* bo   = (const float*)d_in[14];
  float* out = (float*)d_out;

  const int N = in_sizes[0] / 128;
  const int E = in_sizes[1] / 2;
  const int G = out_size / 2;
  const int* src = ei;
  const int* dst = ei + E;

  // workspace layout (all offsets multiples of 4 floats -> 16B aligned)
  float* ws = (float*)d_ws;
  size_t off = 0;
  float* dis    = ws + off; off += (size_t)N;          // degree, then rsqrt
  float* bufA   = ws + off; off += (size_t)N * 128;    // H (x@W), reused per layer
  float* bufB   = ws + off; off += (size_t)N * 128;    // agg -> h (in place)
  float* aArr   = ws + off; off += (size_t)N;
  float* stats  = ws + off; off += 4;
  float* pooled = ws + off; off += (size_t)G * 128;
  float* cnt    = ws + off; off += (size_t)G;
  float* z1     = ws + off; off += (size_t)G * 128;
  float* z2     = ws + off; off += (size_t)G * 128;

  const int NF = N * 128;
  const int GF = G * 128;
  const int mTilesN = (N + 15) / 16;
  const int mTilesG = (G + 15) / 16;

  // --- degree / normalization ---
  zero_kernel<<<1024, 256, 0, stream>>>(dis, N);
  degree_kernel<<<(E + 255) / 256, 256, 0, stream>>>(dst, dis, E);
  rsqrt_kernel<<<(N + 255) / 256, 256, 0, stream>>>(dis, N);

  // --- GCN layer 1 ---
  wmma_gemm_k128<<<mTilesN, 256, 0, stream>>>(x, W1, nullptr, bufA, N, 0);
  zero_kernel<<<4096, 256, 0, stream>>>(bufB, NF);
  mp_scatter<<<8192, 256, 0, stream>>>(bufA, src, dst, dis, bufB, E);
  self_bias_act<<<(NF + 255) / 256, 256, 0, stream>>>(bufB, bufA, dis, b1, NF, 1);

  // --- GCN layer 2 ---
  wmma_gemm_k128<<<mTilesN, 256, 0, stream>>>(bufB, W2, nullptr, bufA, N, 0);
  zero_kernel<<<4096, 256, 0, stream>>>(bufB, NF);
  mp_scatter<<<8192, 256, 0, stream>>>(bufA, src, dst, dis, bufB, E);
  self_bias_act<<<(NF + 255) / 256, 256, 0, stream>>>(bufB, bufA, dis, b2, NF, 0);

  // --- global attention softmax + mean pool ---
  attn_logits<<<(N + 7) / 8, 256, 0, stream>>>(bufB, Wa, ba, aArr, N);
  softmax_stats<<<1, 1024, 0, stream>>>(aArr, stats, N);
  zero_kernel<<<256, 256, 0, stream>>>(pooled, GF);
  zero_kernel<<<4, 256, 0, stream>>>(cnt, G);
  pool_scatter<<<(N + 7) / 8, 256, 0, stream>>>(bufB, aArr, stats, batch, pooled, cnt, N);
  pooled_div<<<(GF + 255) / 256, 256, 0, stream>>>(pooled, cnt, GF);

  // --- MLP head (WMMA) + softmax ---
  wmma_gemm_k128<<<mTilesG, 256, 0, stream>>>(pooled, Wl1, bl1, z1, G, 1);
  wmma_gemm_k128<<<mTilesG, 256, 0, stream>>>(z1, Wl2, bl2, z2, G, 1);
  head_kernel<<<(G + 7) / 8, 256, 0, stream>>>(z2, Wo, bo, out, G);
}